// Net_24850680775390
// MI455X (gfx1250) — compile-verified
//
#include <hip/hip_runtime.h>
#include <hip/hip_bf16.h>
#include <float.h>

// GravNet on MI455X (gfx1250, wave32).
//   kernel 1: project x -> (s, |s|^2) and h
//   kernel 2: brute-force kNN via V_WMMA_F32_16X16X4_F32 ranking metric
//             m = |c|^2 - 2 q.c  (order-equivalent to d^2; |q|^2 added back
//             only for the K=16 survivors). 32 queries per wave (two B frags
//             sharing one A frag), per-lane register top-16 with a v_min-tree
//             guard, double-buffered async global->LDS candidate staging,
//             software-prefetched LDS tile loop. Tail: GravNet aggregation,
//             conv, elu, folded with W_fin into a per-block partial sum.
//   kernel 3: deterministic reduction of block partials (+ b_fin).

#define N_NODES 65536
#define OUT_CH  64
#define CH      512             // candidates per staged chunk (8 KB, x2 buffers)
#define NCH     (N_NODES / CH)  // 128 chunks
#define WAVES   8
#define BLOCK   (WAVES * 32)
#define QPW     32              // queries per wave (2 WMMA query tiles)
#define QPB     (WAVES * QPW)   // 256 queries per block
#define NBLK    (N_NODES / QPB) // 256 blocks

typedef float v2f __attribute__((ext_vector_type(2)));
typedef float v8f __attribute__((ext_vector_type(8)));
typedef int   v4i128 __attribute__((vector_size(16)));   // async-LDS builtin param type

#if __has_builtin(__builtin_amdgcn_global_load_async_to_lds_b128) && \
    __has_builtin(__builtin_amdgcn_s_wait_asynccnt)
#define USE_ASYNC_LDS 1
#endif

__global__ void __launch_bounds__(256)
project_kernel(const float* __restrict__ x,
               const float* __restrict__ Ws, const float* __restrict__ bs,
               const float* __restrict__ Wh, const float* __restrict__ bh,
               float4* __restrict__ s4, float* __restrict__ h)
{
    int i = blockIdx.x * blockDim.x + threadIdx.x;
    if (i >= N_NODES) return;
    float4 xi = ((const float4*)x)[i];
    float sx = bs[0] + Ws[0]*xi.x + Ws[1]*xi.y + Ws[2] *xi.z + Ws[3] *xi.w;
    float sy = bs[1] + Ws[4]*xi.x + Ws[5]*xi.y + Ws[6] *xi.z + Ws[7] *xi.w;
    float sz = bs[2] + Ws[8]*xi.x + Ws[9]*xi.y + Ws[10]*xi.z + Ws[11]*xi.w;
    float hv = bh[0] + Wh[0]*xi.x + Wh[1]*xi.y + Wh[2] *xi.z + Wh[3] *xi.w;
    s4[i] = make_float4(sx, sy, sz, sx*sx + sy*sy + sz*sz);
    h[i]  = hv;
}

__device__ __forceinline__ void insert16(float m, int ci, float bd[16], int bi[16])
{
    float cd = m; int cc = ci;
#pragma unroll
    for (int j = 0; j < 16; ++j) {
        bool sw = cd < bd[j];
        float od = bd[j]; int oi = bi[j];
        bd[j] = sw ? cd : od;  bi[j] = sw ? cc : oi;
        cd    = sw ? od : cd;  cc    = sw ? oi : cc;
    }
}

__global__ void __launch_bounds__(BLOCK)
gravnet_kernel(const float* __restrict__ x,
               const float4* __restrict__ s4,
               const float* __restrict__ h,
               const float* __restrict__ Wout, const float* __restrict__ bout,
               const float* __restrict__ Wfin,
               float* __restrict__ partials)
{
    __shared__ float4 smC[2 * CH];               // double-buffered candidates
    __shared__ float  smMf[WAVES * 32 * 16];     // merge scratch: metrics
    __shared__ int    smMi[WAVES * 32 * 16];     // merge scratch: indices
    __shared__ float  red[BLOCK];                // block reduction

    const int tid   = threadIdx.x;
    const int wv    = tid >> 5;
    const int lane  = tid & 31;
    const int lq    = lane & 15;
    const bool hiL  = lane >= 16;
    const int qbase = blockIdx.x * QPB + wv * QPW;
    const int myq   = qbase + lane;              // query this lane aggregates

    // Two B fragments (4x16 queries each), constant over the candidate sweep.
    // B VGPR0: lanes0-15 K=0 (-2qx), lanes16-31 K=2 (-2qz)
    // B VGPR1: lanes0-15 K=1 (-2qy), lanes16-31 K=3 (1.0)
    float4 sqA = s4[qbase + lq];        // queries qbase+0..15
    float4 sqB = s4[qbase + 16 + lq];   // queries qbase+16..31
    v2f bf0, bf1;
    bf0.x = hiL ? (-2.0f * sqA.z) : (-2.0f * sqA.x);
    bf0.y = hiL ? 1.0f            : (-2.0f * sqA.y);
    bf1.x = hiL ? (-2.0f * sqB.z) : (-2.0f * sqB.x);
    bf1.y = hiL ? 1.0f            : (-2.0f * sqB.y);
    const float myqn = hiL ? sqB.w : sqA.w;      // |q|^2 of this lane's query

    // Per-lane sorted (ascending) top-16 lists: A = query tile 0, B = tile 1.
    float bdA[16], bdB[16];
    int   biA[16], biB[16];
#pragma unroll
    for (int j = 0; j < 16; ++j) {
        bdA[j] = FLT_MAX; biA[j] = 0;
        bdB[j] = FLT_MAX; biB[j] = 0;
    }

    // ---- prologue: stage chunk 0 into buffer 0 ----
#ifdef USE_ASYNC_LDS
#pragma unroll
    for (int kk = 0; kk < CH / BLOCK; ++kk) {
        int ic = tid + kk * BLOCK;
        __builtin_amdgcn_global_load_async_to_lds_b128(
            (__attribute__((address_space(1))) v4i128*)(void*)(s4 + ic),
            (__attribute__((address_space(3))) v4i128*)(void*)(&smC[ic]),
            0, 0);
    }
    __builtin_amdgcn_s_wait_asynccnt(0);
#else
#pragma unroll
    for (int kk = 0; kk < CH / BLOCK; ++kk) {
        int ic = tid + kk * BLOCK;
        smC[ic] = s4[ic];
    }
#endif
    __syncthreads();

    for (int cidx = 0; cidx < NCH; ++cidx) {
        const float4* cur = smC + (cidx & 1) * CH;
        float4* nxt = smC + ((cidx + 1) & 1) * CH;

        // Kick off DMA for the next chunk; overlaps with compute below.
        if (cidx + 1 < NCH) {
            const float4* gsrc = s4 + (cidx + 1) * CH;
#ifdef USE_ASYNC_LDS
#pragma unroll
            for (int kk = 0; kk < CH / BLOCK; ++kk) {
                int ic = tid + kk * BLOCK;
                __builtin_amdgcn_global_load_async_to_lds_b128(
                    (__attribute__((address_space(1))) v4i128*)(void*)(gsrc + ic),
                    (__attribute__((address_space(3))) v4i128*)(void*)(&nxt[ic]),
                    0, 0);
            }
#else
#pragma unroll
            for (int kk = 0; kk < CH / BLOCK; ++kk) {
                int ic = tid + kk * BLOCK;
                nxt[ic] = gsrc[ic];
            }
#endif
        }

        // ---- hot loop: 32 candidate tiles, software-prefetched ----
        float4 c = cur[lq];
        for (int t = 0; t < CH; t += 16) {
            float4 cn = cur[((t + 16) & (CH - 1)) + lq];   // prefetch next tile

            // A fragment (16x4 candidates), shared by both WMMAs:
            // lanes0-15:  VGPR0=cx, VGPR1=cy    (K=0,1; M=lane)
            // lanes16-31: VGPR0=cz, VGPR1=|c|^2 (K=2,3; M=lane-16)
            v2f a;
            a.x = hiL ? c.z : c.x;
            a.y = hiL ? c.w : c.y;

            v8f z = {};
            v8f acc0 = __builtin_amdgcn_wmma_f32_16x16x4_f32(
                false, a, false, bf0, (short)0, z, false, false);
            v8f acc1 = __builtin_amdgcn_wmma_f32_16x16x4_f32(
                false, a, false, bf1, (short)0, z, false, false);

            // D VGPR r: lanes0-15 -> (cand=t+r,   query=lq of tile)
            //           lanes16-31-> (cand=t+r+8, query=lq of tile)
            int cb = cidx * CH + t + (hiL ? 8 : 0);

            float mn0 = fminf(fminf(fminf(acc0[0], acc0[1]), fminf(acc0[2], acc0[3])),
                              fminf(fminf(acc0[4], acc0[5]), fminf(acc0[6], acc0[7])));
            float mn1 = fminf(fminf(fminf(acc1[0], acc1[1]), fminf(acc1[2], acc1[3])),
                              fminf(fminf(acc1[4], acc1[5]), fminf(acc1[6], acc1[7])));

            if (mn0 < bdA[15]) {                       // rare after warmup
#pragma unroll
                for (int r = 0; r < 8; ++r)
                    if (acc0[r] < bdA[15]) insert16(acc0[r], cb + r, bdA, biA);
            }
            if (mn1 < bdB[15]) {
#pragma unroll
                for (int r = 0; r < 8; ++r)
                    if (acc1[r] < bdB[15]) insert16(acc1[r], cb + r, bdB, biB);
            }
            c = cn;
        }

#ifdef USE_ASYNC_LDS
        __builtin_amdgcn_s_wait_asynccnt(0);
#endif
        __syncthreads();   // next chunk visible; everyone done reading cur
    }

    // ---- merge half-lists across lane pairs (lane ^ 16) ----
    // Lane keeps the list of the query it aggregates (lo->tile0, hi->tile1)
    // and donates the other half-list to its partner via LDS.
    float kd[16]; int ki[16];
    const int slot = (wv * 32 + lane) * 16;
#pragma unroll
    for (int j = 0; j < 16; ++j) {
        kd[j] = hiL ? bdB[j] : bdA[j];
        ki[j] = hiL ? biB[j] : biA[j];
        smMf[slot + j] = hiL ? bdA[j] : bdB[j];
        smMi[slot + j] = hiL ? biA[j] : biB[j];
    }
    __syncthreads();
    const int pslot = (wv * 32 + (lane ^ 16)) * 16;
#pragma unroll
    for (int j = 0; j < 16; ++j) {
        float m  = smMf[pslot + j];
        int   ci = smMi[pslot + j];
        if (m < kd[15]) insert16(m, ci, kd, ki);
    }

    // ---- GravNet aggregation: true d^2 = m + |q|^2; w = exp(-10 d^2) ----
    float accm = 0.0f, accx = -FLT_MAX;
#pragma unroll
    for (int j = 0; j < 16; ++j) {
        float d2  = kd[j] + myqn;
        float wgt = __builtin_exp2f(d2 * -14.426950408889634f); // -10*log2(e)
        float msg = h[ki[j]] * wgt;
        accm += msg;
        accx  = fmaxf(accx, msg);
    }
    accm *= (1.0f / 16.0f);

    // conv -> elu -> fold W_fin into one scalar per query (deterministic pool).
    float4 xq = ((const float4*)x)[myq];
    float tq = 0.0f;
#pragma unroll 4
    for (int ch = 0; ch < OUT_CH; ++ch) {
        const float* Wr = Wout + ch * 6;
        float v = bout[ch]
                + Wr[0]*xq.x + Wr[1]*xq.y + Wr[2]*xq.z + Wr[3]*xq.w
                + Wr[4]*accm + Wr[5]*accx;
        float e = v > 0.0f ? v : (__builtin_expf(v) - 1.0f);
        tq += e * Wfin[ch];
    }

    red[tid] = tq;
    __syncthreads();
    for (int s = BLOCK / 2; s > 0; s >>= 1) {
        if (tid < s) red[tid] += red[tid + s];
        __syncthreads();
    }
    if (tid == 0) partials[blockIdx.x] = red[0];
}

__global__ void __launch_bounds__(256)
final_kernel(const float* __restrict__ partials, int nparts,
             const float* __restrict__ bfin, float* __restrict__ out)
{
    __shared__ float red[256];
    int tid = threadIdx.x;
    float s = 0.0f;
    for (int i = tid; i < nparts; i += 256) s += partials[i];
    red[tid] = s;
    __syncthreads();
    for (int st = 128; st > 0; st >>= 1) {
        if (tid < st) red[tid] += red[tid + st];
        __syncthreads();
    }
    if (tid == 0) out[0] = red[0] + bfin[0];
}

extern "C" void kernel_launch(void* const* d_in, const int* in_sizes, int n_in,
                              void* d_out, int out_size, void* d_ws, size_t ws_size,
                              hipStream_t stream)
{
    const float* x    = (const float*)d_in[0];
    const float* Ws   = (const float*)d_in[1];
    const float* bs   = (const float*)d_in[2];
    const float* Wh   = (const float*)d_in[3];
    const float* bh   = (const float*)d_in[4];
    const float* Wout = (const float*)d_in[5];
    const float* bout = (const float*)d_in[6];
    const float* Wfin = (const float*)d_in[7];
    const float* bfin = (const float*)d_in[8];

    char*   ws       = (char*)d_ws;
    float4* s4       = (float4*)ws;                                // N * 16 B
    float*  h        = (float*)(ws + (size_t)N_NODES * 16);        // N * 4 B
    float*  partials = (float*)(ws + (size_t)N_NODES * 20);        // NBLK * 4 B

    project_kernel<<<N_NODES / 256, 256, 0, stream>>>(x, Ws, bs, Wh, bh, s4, h);
    gravnet_kernel<<<NBLK, BLOCK, 0, stream>>>(x, s4, h, Wout, bout, Wfin, partials);
    final_kernel<<<1, 256, 0, stream>>>(partials, NBLK, bfin, (float*)d_out);
}